// SheafConnectionLayer_43662637531917
// MI455X (gfx1250) — compile-verified
//
#include <hip/hip_runtime.h>
#include <hip/hip_bf16.h>

typedef __attribute__((ext_vector_type(2))) float v2f;
typedef __attribute__((ext_vector_type(4))) float v4f;
typedef __attribute__((ext_vector_type(8))) float v8f;

__device__ __forceinline__ float softplusf(float w) {
    // numerically stable softplus
    return fmaxf(w, 0.0f) + log1pf(expf(-fabsf(w)));
}

// out = x  (800000 floats, multiple of 4)
__global__ __launch_bounds__(256) void sheaf_init_kernel(const float* __restrict__ x,
                                                         float* __restrict__ out, int n4) {
    int i = blockIdx.x * blockDim.x + threadIdx.x;
    if (i < n4) {
        ((v4f*)out)[i] = ((const v4f*)x)[i];
    }
}

// One wave (32 lanes) processes 16 edges, both directions, via V_WMMA_F32_16X16X4_F32.
// C rows (M) = edge index in group, C cols (N) = feature dim.
__global__ __launch_bounds__(256) void sheaf_edge_kernel(
    const float* __restrict__ x,     // [N,16]
    const int*   __restrict__ ei,    // [2,E] flat: src=ei[0..E), dst=ei[E..2E)
    const float* __restrict__ deg,   // [N]
    const float* __restrict__ T,     // [2E,16,16] row-major
    const float* __restrict__ rw,    // [2E]
    const float* __restrict__ alphap,// [1]
    float* __restrict__ out,         // [N,16]
    int E)
{
    const int lane  = threadIdx.x & 31;
    const int wid   = threadIdx.x >> 5;
    const int group = blockIdx.x * 8 + wid;
    const int e0    = group * 16;
    if (e0 >= E) return;                      // wave-uniform exit: EXEC stays all-1s below

    const int  n  = lane & 15;                // feature dim handled by this lane (C column)
    const bool hi = lane >= 16;
    const float alpha = *alphap;

    // This lane owns metadata for edge (e0 + n); clamp for (non-occurring) tails.
    const int  el    = e0 + n;
    const bool valid = (el < E);
    const int  elc   = valid ? el : (E - 1);
    const int  s = ei[elc];
    const int  d = ei[E + elc];
    float cf = alpha * softplusf(rw[elc])     / fmaxf(deg[d], 1.0f);
    float cr = alpha * softplusf(rw[E + elc]) / fmaxf(deg[s], 1.0f);
    cf = valid ? cf : 0.0f;
    cr = valid ? cr : 0.0f;

    #pragma unroll
    for (int dir = 0; dir < 2; ++dir) {
        const int   an = (dir == 0) ? s  : d;   // node whose x feeds the A matrix (transformed)
        const int   tn = (dir == 0) ? d  : s;   // node subtracted & scattered to
        const float co = (dir == 0) ? cf : cr;
        const size_t tdb = (size_t)dir * (size_t)E * 256;

        // Preselect this lane's half of its A-source vector:
        // lanes 0-15 supply A[K=4q+0,4q+1], lanes 16-31 supply A[K=4q+2,4q+3] (row = lane&15).
        float pax[4], pay[4];
        {
            const v4f* xa = (const v4f*)(x + (size_t)an * 16);
            #pragma unroll
            for (int q = 0; q < 4; ++q) {
                v4f xc = xa[q];
                pax[q] = hi ? xc.z : xc.x;
                pay[q] = hi ? xc.w : xc.y;
            }
        }

        v8f c = {0.f, 0.f, 0.f, 0.f, 0.f, 0.f, 0.f, 0.f};
        for (int e = 0; e < 16; ++e) {
            const int  ec   = min(e0 + e, E - 1);
            const bool mine = (n == e);
            // B[k, n] = T[ec][n][kk+k]: lane n streams row n of T[ec], 16B at a time, NT hint
            const v4f* Te = (const v4f*)(T + tdb + (size_t)ec * 256 + (size_t)n * 16);
            #pragma unroll
            for (int q = 0; q < 4; ++q) {
                v4f t4 = __builtin_nontemporal_load(&Te[q]);
                v2f a, b;
                a.x = mine ? pax[q] : 0.0f;    // A: only row e nonzero = x_an chunk
                a.y = mine ? pay[q] : 0.0f;
                b.x = hi ? t4.z : t4.x;        // B VGPR0: K=kk (lo lanes) / K=kk+2 (hi lanes)
                b.y = hi ? t4.w : t4.y;        // B VGPR1: K=kk+1 / K=kk+3
                c = __builtin_amdgcn_wmma_f32_16x16x4_f32(
                        false, a, false, b, (short)0, c, false, false);
            }
        }

        // Epilogue: c[r] holds m[edge er][n] with er = r (lanes 0-15) / r+8 (lanes 16-31).
        #pragma unroll
        for (int r = 0; r < 8; ++r) {
            const int   er   = r + (hi ? 8 : 0);
            const int   tn_e = __shfl(tn, er, 32);
            const float co_e = __shfl(co, er, 32);
            const float xsub = x[(size_t)tn_e * 16 + n];
            atomicAdd(out + (size_t)tn_e * 16 + n, co_e * (c[r] - xsub));
        }
    }
}

extern "C" void kernel_launch(void* const* d_in, const int* in_sizes, int n_in,
                              void* d_out, int out_size, void* d_ws, size_t ws_size,
                              hipStream_t stream) {
    const float* x      = (const float*)d_in[0];
    const int*   ei     = (const int*)  d_in[1];
    const float* deg    = (const float*)d_in[2];
    const float* T      = (const float*)d_in[3];
    const float* rw     = (const float*)d_in[4];
    const float* alphap = (const float*)d_in[5];
    float* out = (float*)d_out;

    const int E  = in_sizes[1] / 2;       // 400000
    const int n4 = in_sizes[0] / 4;       // 200000 float4s of x/out

    sheaf_init_kernel<<<(n4 + 255) / 256, 256, 0, stream>>>(x, out, n4);

    const int groups = (E + 15) / 16;     // 25000, exact
    const int blocks = (groups + 7) / 8;  // 8 waves (groups) per 256-thread block
    sheaf_edge_kernel<<<blocks, 256, 0, stream>>>(x, ei, deg, T, rw, alphap, out, E);
}